// offset_Loss_sonnet_45131516346380
// MI455X (gfx1250) — compile-verified
//
#include <hip/hip_runtime.h>

// Problem constants (reference: B=2, C=8, W=H=128, 16 instances)
#define BATCH 2
#define CH 8
#define WDIM 128
#define HDIM 128
#define NINST 16
#define NPIX (WDIM * HDIM)          // 16384 pixels per batch image
#define TOTPIX (BATCH * NPIX)       // 32768

typedef float v2f __attribute__((ext_vector_type(2)));
typedef float v8f __attribute__((ext_vector_type(8)));
typedef int   v4i __attribute__((ext_vector_type(4)));

// ---- gfx1250 async global->LDS copy support (guarded) ----------------------
#if defined(__AMDGCN__) && __has_builtin(__builtin_amdgcn_s_wait_asynccnt) && \
    __has_builtin(__builtin_amdgcn_global_load_async_to_lds_b128)
#define HAVE_ASYNC_LDS 1
#endif

// ---------------------------------------------------------------------------
// Kernel 0: zero the output (loss scalar + gt_sonnet). gt is one-hot per pixel
// over channels (instances are disjoint), so the EDT kernel only writes 1.0
// where its instance owns the pixel; everything else must start at 0.
// ---------------------------------------------------------------------------
__global__ void zero_out_kernel(float* __restrict__ out, int n) {
  for (int i = blockIdx.x * blockDim.x + threadIdx.x; i < n;
       i += gridDim.x * blockDim.x)
    out[i] = 0.0f;
}

// ---------------------------------------------------------------------------
// Kernel 1: per-(b,k) exact EDT + normalize + threshold + bin -> gt_sonnet.
// One workgroup per (batch, instance). Whole tile resident in LDS:
//   g2 : 32 KB u16 — doubles as async staging buffer for the mask tile,
//        then vertical squared distances, overwritten in place by D2
//   nb :  2 KB bits — nucleus bitmask built with wave32 ballot
// ---------------------------------------------------------------------------
__global__ __launch_bounds__(256) void edt_gt_kernel(const int* __restrict__ mask,
                                                     float* __restrict__ gt) {
  __shared__ unsigned short g2[NPIX];   // 32 KB
  __shared__ unsigned nb[WDIM][4];      // 128x128 bits = 2 KB
  __shared__ float red[256];

  const int k = blockIdx.x;             // instance 0..15 (label k+1)
  const int b = blockIdx.y;             // batch
  const int tid = threadIdx.x;
  const int lane = tid & 31;
  const int label = k + 1;
  const int* mb = mask + b * NPIX;

  // Phase A: nucleus bitmask via wave32 ballot (lanes cover 32 consecutive j)
#ifdef HAVE_ASYNC_LDS
  // Stage the 64 KB i32 mask tile through LDS in two 32 KB halves using the
  // async copy path (no VGPR round-trip, tracked on ASYNCcnt), reusing g2.
  int* stage = (int*)g2;                          // 8192 ints = 32 KB
  for (int half = 0; half < 2; ++half) {
    const int base = half * 8192;                 // int offset into mask tile
    for (int t = tid; t < 8192 / 4; t += 256)     // 16 B per lane per op
      __builtin_amdgcn_global_load_async_to_lds_b128(
          (v4i*)(mb + base + t * 4), (v4i*)(stage + t * 4), 0, 0);
    __builtin_amdgcn_s_wait_asynccnt(0);
    __syncthreads();                              // all waves' copies landed
    for (int t = tid; t < 8192; t += 256) {
      const int p = base + t;
      bool nuc = (stage[t] == label);
#if __has_builtin(__builtin_amdgcn_ballot_w32)
      unsigned bal = __builtin_amdgcn_ballot_w32(nuc);
#else
      unsigned bal = (unsigned)__ballot(nuc);
#endif
      if (lane == 0) nb[p >> 7][(p & 127) >> 5] = bal;
    }
    __syncthreads();                              // buffer dead: safe to reuse
  }
#else
  for (int p = tid; p < NPIX; p += 256) {
    bool nuc = (mb[p] == label);
#if __has_builtin(__builtin_amdgcn_ballot_w32)
    unsigned bal = __builtin_amdgcn_ballot_w32(nuc);
#else
    unsigned bal = (unsigned)__ballot(nuc);
#endif
    if (lane == 0) nb[p >> 7][(p & 127) >> 5] = bal;
  }
  __syncthreads();
#endif

  // Phase B: per-column 1D EDT (distance to nearest background row), O(W).
  if (tid < HDIM) {
    const int j = tid;
    int df = 1 << 20;
    for (int i = 0; i < WDIM; ++i) {
      bool bg = !((nb[i][j >> 5] >> (j & 31)) & 1u);
      df = bg ? 0 : df + 1;
      g2[i * HDIM + j] = (unsigned short)(df > 65535 ? 65535 : df);
    }
    int db = 1 << 20;
    for (int i = WDIM - 1; i >= 0; --i) {
      bool bg = !((nb[i][j >> 5] >> (j & 31)) & 1u);
      db = bg ? 0 : db + 1;
      int d1 = (int)g2[i * HDIM + j];
      if (db < d1) d1 = db;
      g2[i * HDIM + j] =
          (d1 > 127) ? (unsigned short)0xFFFF : (unsigned short)(d1 * d1);
    }
  }
  __syncthreads();

  // Phase C: horizontal min-plus  D2[i,j] = min_j' (G2[i,j'] + (j-j')^2),
  // done in place two rows at a time (256 threads = 2 rows x 128 cols).
  for (int rb = 0; rb < WDIM; rb += 2) {
    const int i = rb + (tid >> 7);
    const int j = tid & 127;
    const unsigned short* grow = &g2[i * HDIM];
    unsigned best = 0xFFFFFFFFu;
    for (int jp = 0; jp < HDIM; ++jp) {
      int dj = j - jp;
      unsigned t = (unsigned)grow[jp] + (unsigned)(dj * dj);
      if (t < best) best = t;
    }
    __syncthreads();                       // all reads of rows rb, rb+1 done
    g2[i * HDIM + j] =
        (best >= 0xFFFFu) ? (unsigned short)0xFFFF : (unsigned short)best;
    __syncthreads();
  }

  // Phase D: tile max of d = sqrt(D2)
  float lmax = 0.0f;
  for (int p = tid; p < NPIX; p += 256) {
    unsigned short d2 = g2[p];
    float d = (d2 == 0xFFFFu) ? 31622.7766f : sqrtf((float)d2);
    lmax = fmaxf(lmax, d);
  }
  red[tid] = lmax;
  __syncthreads();
  for (int s = 128; s > 0; s >>= 1) {
    if (tid < s) red[tid] = fmaxf(red[tid], red[tid + s]);
    __syncthreads();
  }
  const float m = red[0];
  const float inv_m = (m > 0.0f) ? (1.0f / m) : 1.0f;

  // Phase E: normalize, threshold, bin into one of 8 channels (one-hot).
  const float ddd[7] = {0.83f, 0.68f, 0.54f, 0.41f, 0.29f, 0.18f, 0.09f};
  for (int p = tid; p < NPIX; p += 256) {
    const int i = p >> 7, j = p & 127;
    if ((nb[i][j >> 5] >> (j & 31)) & 1u) {
      unsigned short d2 = g2[p];
      float d = (d2 == 0xFFFFu) ? 31622.7766f : sqrtf((float)d2);
      float dn = d * inv_m;
      if (dn < 0.5f) dn = 0.0f;
      if (dn > 0.7f) dn = 1.0f;
      int c = 0;                 // first channel with dn >= ddd[c] (half-open bins)
      #pragma unroll
      for (int cc = 0; cc < 7; ++cc) c += (dn < ddd[cc]) ? 1 : 0;
      gt[((b * CH + c) * NPIX) + p] = 1.0f;
    }
  }
}

// ---------------------------------------------------------------------------
// Kernel 2: smooth-L1 loss, channel-mean, ignored-map weight, global mean.
// The sum reduction is mapped onto V_WMMA_F32_16X16X4_F32 (exact f32):
// each WMMA folds 64 per-pixel loss values (2 per lane, A 16x4) against an
// all-ones B into the 16x16 accumulator; every accumulator column is an
// identical row-sum vector, so the grand total is (sum of all 256)/16.
// ---------------------------------------------------------------------------
__global__ __launch_bounds__(256) void loss_kernel(const float* __restrict__ pred,
                                                   const float* __restrict__ ig,
                                                   const float* __restrict__ gt,
                                                   float* __restrict__ out) {
  const int tid = threadIdx.x;
  const int lane = tid & 31;
  const int wave = blockIdx.x * (blockDim.x >> 5) + (tid >> 5);
  const int nwaves = gridDim.x * (blockDim.x >> 5);
  const int nchunks = TOTPIX / 64;       // 512 chunks of 64 pixels

  v8f acc = {};
  v2f bones; bones.x = 1.0f; bones.y = 1.0f;

  for (int chunk = wave; chunk < nchunks; chunk += nwaves) {
    v2f a;
    #pragma unroll
    for (int q = 0; q < 2; ++q) {
      const int p = chunk * 64 + lane * 2 + q;      // always < TOTPIX
      const int bb = p >> 14;                       // p / NPIX
      const int pix = p & (NPIX - 1);
      float s = 0.0f;
      #pragma unroll
      for (int c = 0; c < CH; ++c) {
        const int idx = ((bb * CH + c) * NPIX) + pix;
        const float diff = pred[idx] - gt[idx];
        const float ad = fabsf(diff);
        s += (ad < 1.0f) ? 0.5f * diff * diff : (ad - 0.5f);
      }
      const float v = s * 0.125f * ig[p];
      if (q == 0) a.x = v; else a.y = v;
    }
#if __has_builtin(__builtin_amdgcn_wmma_f32_16x16x4_f32)
    acc = __builtin_amdgcn_wmma_f32_16x16x4_f32(
        /*neg_a=*/false, a, /*neg_b=*/false, bones,
        /*c_mod=*/(short)0, acc, /*reuse_a=*/false, /*reuse_b=*/false);
#else
    acc[0] += a.x + a.y;   // scalar fallback (histogram will show wmma==0)
#endif
  }

  float s = acc[0] + acc[1] + acc[2] + acc[3] + acc[4] + acc[5] + acc[6] + acc[7];
  #pragma unroll
  for (int off = 16; off >= 1; off >>= 1) s += __shfl_xor(s, off, 32);
#if __has_builtin(__builtin_amdgcn_wmma_f32_16x16x4_f32)
  const float scale = (1.0f / 16.0f) * (1.0f / (float)TOTPIX);
#else
  const float scale = 1.0f / (float)TOTPIX;
#endif
  if (lane == 0) atomicAdd(out, s * scale);
}

// ---------------------------------------------------------------------------
extern "C" void kernel_launch(void* const* d_in, const int* in_sizes, int n_in,
                              void* d_out, int out_size, void* d_ws, size_t ws_size,
                              hipStream_t stream) {
  (void)in_sizes; (void)n_in; (void)out_size; (void)d_ws; (void)ws_size;
  const float* pred = (const float*)d_in[0];   // [2,8,128,128] f32
  const int*   mask = (const int*)d_in[1];     // [2,128,128]   i32
  const float* ig   = (const float*)d_in[2];   // [2,128,128]   f32
  float* out = (float*)d_out;                  // [0]=loss, [1..]=gt_sonnet
  float* gt  = out + 1;

  zero_out_kernel<<<64, 256, 0, stream>>>(out, 1 + BATCH * CH * NPIX);
  edt_gt_kernel<<<dim3(NINST, BATCH), 256, 0, stream>>>(mask, gt);
  loss_kernel<<<16, 256, 0, stream>>>(pred, ig, gt, out);
}